// MemoryLayer_41472204210702
// MI455X (gfx1250) — compile-verified
//
#include <hip/hip_runtime.h>
#include <math.h>

#define T_SEQ 2048
#define DMODEL 256

typedef __attribute__((ext_vector_type(2))) float v2f;
typedef __attribute__((ext_vector_type(4))) float v4f;
typedef __attribute__((ext_vector_type(8))) float v8f;

// ---------------------------------------------------------------------------
// Kernel 1: keys_raw = X@Wk + bk, values = X@Wv + bv via V_WMMA_F32_16X16X4_F32
// One wave per 16x16 output tile; K-loop in steps of 4.
// A-frag layout (32-bit A 16x4, ISA 7.12.2): lane<16 -> (K=k0,k0+1) of row M=lane;
// lane>=16 -> (K=k0+2,k0+3) of row M=lane-16. B mirrors (rows K striped).
// C/D: VGPR m -> row M = m + 8*(lane>=16), col N = lane&15.
// ---------------------------------------------------------------------------
__global__ __launch_bounds__(256) void proj_wmma_kernel(
    const float* __restrict__ X, const float* __restrict__ Wk, const float* __restrict__ bk,
    const float* __restrict__ Wv, const float* __restrict__ bv,
    float* __restrict__ keys, float* __restrict__ vals)
{
  const int lane = threadIdx.x & 31;
  const int wave = threadIdx.x >> 5;
  const int tile = blockIdx.x * 8 + wave;     // 2048 tiles = (2048/16)*(256/16)
  const int tn = tile & 15;                   // 16 col tiles
  const int tm = tile >> 4;                   // 128 row tiles
  const int r0 = tm * 16, n0 = tn * 16;
  const int l15 = lane & 15;
  const int hi  = lane >> 4;                  // 0 or 1

  v8f acck = {}; v8f accv = {};
  const float* arow = X + (size_t)(r0 + l15) * DMODEL + 2 * hi;
  #pragma unroll 4
  for (int k0 = 0; k0 < DMODEL; k0 += 4) {
    v2f a = *(const v2f*)(arow + k0);
    const float* bpk = Wk + (size_t)(k0 + 2 * hi) * DMODEL + n0 + l15;
    const float* bpv = Wv + (size_t)(k0 + 2 * hi) * DMODEL + n0 + l15;
    v2f bfk; bfk.x = bpk[0]; bfk.y = bpk[DMODEL];
    v2f bfv; bfv.x = bpv[0]; bfv.y = bpv[DMODEL];
    acck = __builtin_amdgcn_wmma_f32_16x16x4_f32(false, a, false, bfk, (short)0, acck, false, false);
    accv = __builtin_amdgcn_wmma_f32_16x16x4_f32(false, a, false, bfv, (short)0, accv, false, false);
  }
  const float biask = bk[n0 + l15];
  const float biasv = bv[n0 + l15];
  #pragma unroll
  for (int m = 0; m < 8; ++m) {
    const int r = r0 + m + 8 * hi;
    keys[(size_t)r * DMODEL + n0 + l15] = acck[m] + biask;
    vals[(size_t)r * DMODEL + n0 + l15] = accv[m] + biasv;
  }
}

// ---------------------------------------------------------------------------
// Kernel 2: row-normalize keys (in place) + beta/gamma = sigmoid(X@Wb/Wg + b).
// One wave32 per sequence position t; shfl_xor wave reductions.
// ---------------------------------------------------------------------------
__global__ __launch_bounds__(256) void rowprep_kernel(
    const float* __restrict__ X, float* __restrict__ keys,
    const float* __restrict__ Wb, const float* __restrict__ bb,
    const float* __restrict__ Wg, const float* __restrict__ bg,
    float* __restrict__ beta, float* __restrict__ gamma)
{
  const int lane = threadIdx.x & 31;
  const int wave = threadIdx.x >> 5;
  const int t = blockIdx.x * 8 + wave;        // 2048 rows
  float kreg[8];
  float ss = 0.f, db = 0.f, dg = 0.f;
  #pragma unroll
  for (int m = 0; m < 8; ++m) {
    const int j = lane + 32 * m;
    const float kv = keys[(size_t)t * DMODEL + j];
    kreg[m] = kv; ss += kv * kv;
    const float xv = X[(size_t)t * DMODEL + j];
    db += xv * Wb[j];
    dg += xv * Wg[j];
  }
  #pragma unroll
  for (int off = 16; off >= 1; off >>= 1) {
    ss += __shfl_xor(ss, off, 32);
    db += __shfl_xor(db, off, 32);
    dg += __shfl_xor(dg, off, 32);
  }
  const float inv = 1.0f / sqrtf(ss);
  #pragma unroll
  for (int m = 0; m < 8; ++m)
    keys[(size_t)t * DMODEL + lane + 32 * m] = kreg[m] * inv;
  if (lane == 0) {
    beta[t]  = 1.0f / (1.0f + expf(-(db + bb[0])));
    gamma[t] = 1.0f / (1.0f + expf(-(dg + bg[0])));
  }
}

// ---------------------------------------------------------------------------
// Kernel 3: sequential delta-rule scan. Rows of c are independent:
// one wave32 per row (blocks 0..255); each lane owns 8 CONTIGUOUS columns
// [lane*8, lane*8+8) so state loads, k-prefetch and the streaming output
// stores are all 128-bit ops (512 B per wave-instruction, fully coalesced).
// Software-pipelined register prefetch of k_{t+1}; non-temporal stores
// (c_seq is 512 MB write-once -> the HBM-store roofline dominates).
// Blocks 256..263 handle the 256-wide s scan.
// ---------------------------------------------------------------------------
__global__ __launch_bounds__(32) void scan_kernel(
    const float* __restrict__ keys, const float* __restrict__ vals,
    const float* __restrict__ beta, const float* __restrict__ gamma,
    const float* __restrict__ c_prev, const float* __restrict__ s_prev,
    float* __restrict__ c_out, float* __restrict__ s_out)
{
  const int lane = threadIdx.x;               // 0..31
  if (blockIdx.x < DMODEL) {
    const int row = blockIdx.x;
    const int col0 = lane * 8;                // 8 contiguous columns per lane
    v4f c0 = *(const v4f*)(c_prev + (size_t)row * DMODEL + col0);
    v4f c1 = *(const v4f*)(c_prev + (size_t)row * DMODEL + col0 + 4);
    v4f k0 = *(const v4f*)(keys + col0);
    v4f k1 = *(const v4f*)(keys + col0 + 4);
    for (int t = 0; t < T_SEQ; ++t) {
      // prefetch next step's key row into registers (overlaps with compute)
      const int tp = (t + 1 < T_SEQ) ? (t + 1) : t;
      const v4f kn0 = *(const v4f*)(keys + (size_t)tp * DMODEL + col0);
      const v4f kn1 = *(const v4f*)(keys + (size_t)tp * DMODEL + col0 + 4);
      const float g = gamma[t];
      const float b = beta[t];
      const float v = vals[(size_t)t * DMODEL + row];
      // vbar_row = dot(c[row,:], k_t)
      float partial = c0.x * k0.x + c0.y * k0.y + c0.z * k0.z + c0.w * k0.w
                    + c1.x * k1.x + c1.y * k1.y + c1.z * k1.z + c1.w * k1.w;
      #pragma unroll
      for (int off = 16; off >= 1; off >>= 1)
        partial += __shfl_xor(partial, off, 32);
      const float scale = b * g * (v - partial);   // b*g*new_v[row]
      const float decay = 1.0f - g;
      c0 = c0 * decay + k0 * scale;
      c1 = c1 * decay + k1 * scale;
      float* dst = c_out + ((size_t)t * DMODEL + row) * DMODEL + col0;
      __builtin_nontemporal_store(c0, (v4f*)dst);
      __builtin_nontemporal_store(c1, (v4f*)(dst + 4));
      k0 = kn0; k1 = kn1;
    }
  } else {
    const int j = (blockIdx.x - DMODEL) * 32 + lane;   // 0..255
    float s_state = s_prev[j];
    for (int t = 0; t < T_SEQ; ++t) {
      const float g = gamma[t];
      const float b = beta[t];
      const float k = keys[(size_t)t * DMODEL + j];
      s_state = s_state * (1.0f - g) + k * g * b;
      __builtin_nontemporal_store(s_state, &s_out[(size_t)t * DMODEL + j]);
    }
  }
}

extern "C" void kernel_launch(void* const* d_in, const int* in_sizes, int n_in,
                              void* d_out, int out_size, void* d_ws, size_t ws_size,
                              hipStream_t stream) {
  const float* X      = (const float*)d_in[0];
  const float* c_prev = (const float*)d_in[1];
  const float* s_prev = (const float*)d_in[2];
  const float* Wk     = (const float*)d_in[3];
  const float* bk     = (const float*)d_in[4];
  const float* Wv     = (const float*)d_in[5];
  const float* bv     = (const float*)d_in[6];
  const float* Wb     = (const float*)d_in[7];
  const float* bb     = (const float*)d_in[8];
  const float* Wg     = (const float*)d_in[9];
  const float* bg     = (const float*)d_in[10];

  float* c_out = (float*)d_out;                               // [T,D,D]
  float* s_out = c_out + (size_t)T_SEQ * DMODEL * DMODEL;     // [T,D]

  float* ws    = (float*)d_ws;
  float* keys  = ws;                                          // [T,D] raw -> normalized
  float* vals  = keys + (size_t)T_SEQ * DMODEL;               // [T,D]
  float* beta  = vals + (size_t)T_SEQ * DMODEL;               // [T]
  float* gamma = beta + T_SEQ;                                // [T]

  proj_wmma_kernel<<<256, 256, 0, stream>>>(X, Wk, bk, Wv, bv, keys, vals);
  rowprep_kernel<<<256, 256, 0, stream>>>(X, keys, Wb, bb, Wg, bg, beta, gamma);
  scan_kernel<<<DMODEL + 8, 32, 0, stream>>>(keys, vals, beta, gamma,
                                             c_prev, s_prev, c_out, s_out);
}